// TXCBareAntidead_51719996178883
// MI455X (gfx1250) — compile-verified
//
#include <hip/hip_runtime.h>
#include <hip/hip_bf16.h>

// ---------------- problem constants ----------------
constexpr int BN   = 4096;           // batch rows
constexpr int TT   = 4;
constexpr int DIN  = 256;
constexpr int KD   = TT * DIN;       // 1024 (GEMM K)
constexpr int SN   = 24576;          // d_sae (GEMM N)
constexpr int TOPK = 128;
constexpr int AUXK = 512;
constexpr int DEAD_THRESHOLD = 10000000;

// ---------------- workspace layout (in 4-byte words) ----------------
constexpr size_t OFF_SCAL   = 0;         // [0]=recon_sum f32, [1]=l2a_sum, [2]=denom_sum, [3]=n_dead (int)
constexpr size_t OFF_MUSUM  = 16;        // 256 f32
constexpr size_t OFF_ACTIVE = 272;       // 24576 int
constexpr size_t ZERO_WORDS = OFF_ACTIVE + SN;            // 24848 (zeroed every call)
constexpr size_t OFF_DEAD   = ZERO_WORDS;                 // 24576 int
constexpr size_t OFF_CNTM   = OFF_DEAD + SN;              // 4096 int
constexpr size_t OFF_CNTA   = OFF_CNTM + BN;              // 4096 int
constexpr size_t OFF_IDXM   = OFF_CNTA + BN;              // 4096*128 int
constexpr size_t OFF_VALM   = OFF_IDXM + (size_t)BN*TOPK; // 4096*128 f32
constexpr size_t OFF_IDXA   = OFF_VALM + (size_t)BN*TOPK; // 4096*512 int
constexpr size_t OFF_VALA   = OFF_IDXA + (size_t)BN*AUXK; // 4096*512 f32
constexpr size_t OFF_RES    = OFF_VALA + (size_t)BN*AUXK; // 4096*1024 f32 residual

// ---------------- CDNA5 WMMA types ----------------
typedef __attribute__((ext_vector_type(2))) float v2f;
typedef __attribute__((ext_vector_type(8))) float v8f;
typedef __attribute__((__vector_size__(4 * sizeof(int)))) int gv4i;  // matches builtin param type

// optional async global->LDS path (gfx1250)
#if defined(__has_builtin)
#if __has_builtin(__builtin_amdgcn_global_load_async_to_lds_b128) && \
    __has_builtin(__builtin_amdgcn_s_wait_asynccnt)
#define TXC_ASYNC 1
#endif
#endif
#ifndef TXC_ASYNC
#define TXC_ASYNC 0
#endif

// ============================================================
// K0: zero the accumulator region of the workspace
// ============================================================
__global__ void k_zero(float* __restrict__ ws, int n) {
  int i = blockIdx.x * 256 + threadIdx.x;
  if (i < n) ws[i] = 0.0f;
}

// ============================================================
// K1: encoder GEMM  pre[m][n] = sum_k x[m][k]*Wenc[k][n] + benc[n]
// fp32 WMMA 16x16x4, 64x64 block tile, 8 waves (4Mx2N), Kc=16 via LDS.
// A staged [m][k] (row stride 20 -> 16B-aligned rows): fragment = one
// aligned ds_load_b64 into an even VGPR pair. B staged [k][n] so the
// async global->LDS b128 copy is a straight contiguous transfer.
// ============================================================
__global__ __launch_bounds__(256) void k_encode(const float* __restrict__ x,
                                                const float* __restrict__ Wenc,
                                                const float* __restrict__ benc,
                                                float* __restrict__ pre) {
  __shared__ __align__(16) float As[64][20];  // [m][k], +4 pad keeps float4/v2f alignment
  __shared__ __align__(16) float Bs[16][68];  // [k][n]
  const int tid  = threadIdx.x;
  const int n0   = blockIdx.x * 64;
  const int m0   = blockIdx.y * 64;
  const int lane = tid & 31;
  const int wid  = tid >> 5;
  const int wm   = wid & 3;        // M sub-tile (16 rows)
  const int wn   = wid >> 2;       // N sub-tile (32 cols -> two 16x16 tiles)
  const int h    = lane >> 4;      // lane half
  const int l16  = lane & 15;

  v8f c0 = {0, 0, 0, 0, 0, 0, 0, 0};
  v8f c1 = {0, 0, 0, 0, 0, 0, 0, 0};

  const int ar = tid >> 2, ac = (tid & 3) << 2;   // A: 64 rows x 16 k
  const int br = tid >> 4, bc = (tid & 15) << 2;  // B: 16 k x 64 n
  const int am = wm * 16 + l16;                   // this lane's A row

  for (int k0 = 0; k0 < KD; k0 += 16) {
    // B tile: contiguous rows of W_enc (k-major, stride SN)
#if TXC_ASYNC
    __builtin_amdgcn_global_load_async_to_lds_b128(
        (gv4i*)(Wenc + (size_t)(k0 + br) * SN + n0 + bc),
        (gv4i*)&Bs[br][bc], 0, 0);
#else
    float4 bv = *(const float4*)(Wenc + (size_t)(k0 + br) * SN + n0 + bc);
    *(float4*)&Bs[br][bc] = bv;
#endif
    // A tile: natural [m][k] layout, one aligned float4 store
    float4 av = *(const float4*)(x + (size_t)(m0 + ar) * KD + k0 + ac);
    *(float4*)&As[ar][ac] = av;
#if TXC_ASYNC
    __builtin_amdgcn_s_wait_asynccnt(0);
#endif
    __syncthreads();

#pragma unroll
    for (int kk = 0; kk < 16; kk += 4) {
      const int ka = kk + 2 * h;  // f32 A/B layout: VGPR0 holds K=0 (lanes 0-15) / K=2 (16-31)
      const v2f a = *(const v2f*)&As[am][ka];  // ds_load_b64, even-aligned pair
      v2f b0, b1;
      b0.x = Bs[ka][wn * 32 + l16];
      b0.y = Bs[ka + 1][wn * 32 + l16];
      b1.x = Bs[ka][wn * 32 + 16 + l16];
      b1.y = Bs[ka + 1][wn * 32 + 16 + l16];
      c0 = __builtin_amdgcn_wmma_f32_16x16x4_f32(false, a, false, b0, (short)0, c0, false, false);
      c1 = __builtin_amdgcn_wmma_f32_16x16x4_f32(false, a, false, b1, (short)0, c1, false, false);
    }
    __syncthreads();
  }

  // epilogue: C layout VGPR p -> M=p (lanes 0-15) / M=p+8 (lanes 16-31), N=lane%16
  const int gn = n0 + wn * 32 + l16;
  const float be0 = benc[gn];
  const float be1 = benc[gn + 16];
#pragma unroll
  for (int p = 0; p < 8; ++p) {
    const int gm = m0 + wm * 16 + p + 8 * h;
    pre[(size_t)gm * SN + gn]      = c0[p] + be0;
    pre[(size_t)gm * SN + gn + 16] = c1[p] + be1;
  }
}

// ---------------- helpers ----------------
__device__ inline unsigned mapf(float f) {
  unsigned u = __float_as_uint(f);
  return (u & 0x80000000u) ? ~u : (u | 0x80000000u);  // monotone order-preserving map
}

// MSD radix select: bit pattern of the Kwant-th largest value + #equals to keep
__device__ void radix_select(const float* __restrict__ rowp, const int* __restrict__ dead,
                             int Kwant, int tid, int* hist, unsigned& thresh, int& budget) {
  unsigned prefix = 0;
  int remaining = Kwant;
  for (int shift = 24; shift >= 0; shift -= 8) {
    hist[tid] = 0;
    __syncthreads();
    const unsigned hmask = (shift == 24) ? 0u : (0xFFFFFFFFu << (shift + 8));
    for (int i = tid; i < SN; i += 256) {
      float v = rowp[i];
      if (dead) v = dead[i] ? fmaxf(v, 0.0f) : 0.0f;
      const unsigned u = mapf(v);
      if ((u & hmask) == (prefix & hmask)) atomicAdd(&hist[(u >> shift) & 255], 1);
    }
    __syncthreads();
    int rem = remaining, digit = 0;
    for (int d = 255; d >= 0; --d) {
      const int c = hist[d];
      if (c >= rem) { digit = d; break; }
      rem -= c;
    }
    prefix |= ((unsigned)digit) << shift;
    remaining = rem;
    __syncthreads();
  }
  thresh = prefix;
  budget = remaining;
}

// ordered block-wide scan of a predicate (ballot + per-wave totals), wave32-safe
__device__ inline void scanBool(bool pred, int tid, volatile int* wsum, int& excl, int& tot) {
  const unsigned mask = (unsigned)__ballot(pred);
  const int lane = tid & 31, wv = tid >> 5;
  const int lp = __popc(mask & ((1u << lane) - 1u));
  if (lane == 0) wsum[wv] = __popc(mask);
  __syncthreads();
  int off = 0, t = 0;
#pragma unroll
  for (int w = 0; w < 8; ++w) {
    const int c = wsum[w];
    t += c;
    if (w < wv) off += c;
  }
  excl = off + lp;
  tot = t;
  __syncthreads();
}

__device__ inline float blockReduceSum(float v, volatile float* red, int tid) {
  __syncthreads();
  red[tid] = v;
  __syncthreads();
  for (int s = 128; s > 0; s >>= 1) {
    if (tid < s) red[tid] += red[tid + s];
    __syncthreads();
  }
  return red[0];
}

// ============================================================
// K2: per-row top-128 select; build ordered compact list; mark active
// ============================================================
__global__ __launch_bounds__(256) void k_select_main(const float* __restrict__ pre,
                                                     int* __restrict__ idxM, float* __restrict__ valM,
                                                     int* __restrict__ cntM, int* __restrict__ active) {
  __shared__ int hist[256];
  __shared__ int wsum[8];
  const int tid = threadIdx.x;
  const int row = blockIdx.x;
  const float* rowp = pre + (size_t)row * SN;

  unsigned thresh; int budget;
  radix_select(rowp, nullptr, TOPK, tid, hist, thresh, budget);

  int eqUsed = 0, outCount = 0;
  for (int base = 0; base < SN; base += 256) {
    const int s = base + tid;
    const float v = rowp[s];
    const unsigned u = mapf(v);
    const bool isEq = (u == thresh);
    int ep, et;
    scanBool(isEq, tid, wsum, ep, et);
    const bool sel = (u > thresh) || (isEq && (eqUsed + ep) < budget);
    const bool keep = sel && (v > 0.0f);  // z = relu(vals): only positives matter downstream
    int kp, kt;
    scanBool(keep, tid, wsum, kp, kt);
    if (keep) {
      const size_t p = (size_t)row * TOPK + outCount + kp;
      idxM[p] = s;
      valM[p] = v;
      active[s] = 1;
    }
    eqUsed += et;
    outCount += kt;
  }
  if (tid == 0) cntM[row] = outCount;
}

// ============================================================
// K3: dead-feature mask + n_dead
// ============================================================
__global__ void k_dead(const int* __restrict__ ntsf, const int* __restrict__ active,
                       int* __restrict__ dead, int* __restrict__ ndead) {
  const int s = blockIdx.x * 256 + threadIdx.x;
  if (s < SN) {
    int c = ntsf[s] + BN * TT;
    if (active[s]) c = 0;
    const int d = (c >= DEAD_THRESHOLD) ? 1 : 0;
    dead[s] = d;
    if (d) atomicAdd(ndead, 1);
  }
}

// ============================================================
// K4: per-row aux top-512 on dead-masked relu(pre); build aux list;
//     finalize z in place (zero + scatter main list)
// ============================================================
__global__ __launch_bounds__(256) void k_select_aux(float* __restrict__ z, const int* __restrict__ dead,
                                                    const int* __restrict__ idxM, const float* __restrict__ valM,
                                                    const int* __restrict__ cntM,
                                                    int* __restrict__ idxA, float* __restrict__ valA,
                                                    int* __restrict__ cntA) {
  __shared__ int hist[256];
  __shared__ int wsum[8];
  const int tid = threadIdx.x;
  const int row = blockIdx.x;
  float* rowp = z + (size_t)row * SN;

  unsigned athr; int abud;
  radix_select(rowp, dead, AUXK, tid, hist, athr, abud);

  int eqUsed = 0, outCount = 0;
  for (int base = 0; base < SN; base += 256) {
    const int s = base + tid;
    const float v = rowp[s];
    const float av = dead[s] ? fmaxf(v, 0.0f) : 0.0f;
    const unsigned u = mapf(av);
    const bool isEq = (u == athr);
    int ep, et;
    scanBool(isEq, tid, wsum, ep, et);
    const bool sel = (u > athr) || (isEq && (eqUsed + ep) < abud);
    const bool keep = sel && (av > 0.0f);  // zero-valued selections contribute nothing
    int kp, kt;
    scanBool(keep, tid, wsum, kp, kt);
    if (keep) {
      const size_t p = (size_t)row * AUXK + outCount + kp;
      idxA[p] = s;
      valA[p] = av;
    }
    rowp[s] = 0.0f;  // zero z; main scatter restores selected positives below
    eqUsed += et;
    outCount += kt;
  }
  __syncthreads();
  const int c = cntM[row];
  for (int e = tid; e < c; e += 256)
    rowp[idxM[(size_t)row * TOPK + e]] = valM[(size_t)row * TOPK + e];
  if (tid == 0) cntA[row] = outCount;
}

// ============================================================
// K5: sparse decode -> x_hat, residual  (W_dec is L2-resident: 100MB < 192MB)
// ============================================================
__global__ __launch_bounds__(256) void k_decode(const float* __restrict__ x, const float* __restrict__ Wdec,
                                                const float* __restrict__ bdec,
                                                const int* __restrict__ idxM, const float* __restrict__ valM,
                                                const int* __restrict__ cntM,
                                                float* __restrict__ xhat, float* __restrict__ resid) {
  const int b = blockIdx.x;
  const int tid = threadIdx.x;
  const int d0 = tid * 4;  // thread owns 4 of the 1024 output dims
  float a0 = 0.f, a1 = 0.f, a2 = 0.f, a3 = 0.f;
  const int c = cntM[b];
  const int* ip = idxM + (size_t)b * TOPK;
  const float* vp = valM + (size_t)b * TOPK;
  for (int e = 0; e < c; ++e) {
    const int id = ip[e];        // uniform -> scalar load
    const float vv = vp[e];
    const float4 w = *(const float4*)(Wdec + (size_t)id * KD + d0);
    a0 += vv * w.x; a1 += vv * w.y; a2 += vv * w.z; a3 += vv * w.w;
  }
  const size_t o = (size_t)b * KD + d0;
  const float acc[4] = {a0, a1, a2, a3};
#pragma unroll
  for (int j = 0; j < 4; ++j) {
    const float xh = acc[j] + bdec[d0 + j];
    xhat[o + j] = xh;
    resid[o + j] = x[o + j] - xh;
  }
}

// ============================================================
// K6: column sums of residual (mu) + total sum of residual^2 (recon)
// ============================================================
__global__ __launch_bounds__(256) void k_mu(const float* __restrict__ resid,
                                            float* __restrict__ musum, float* __restrict__ scal) {
  __shared__ float red[256];
  const int tid = threadIdx.x;
  const int r0 = blockIdx.x * 256;  // 64 blocks x 256 (b,t)-rows
  float ms = 0.f, sq = 0.f;
  for (int r = 0; r < 256; ++r) {
    const float v = resid[(size_t)(r0 + r) * DIN + tid];  // thread owns d_in column = tid
    ms += v;
    sq += v * v;
  }
  atomicAdd(&musum[tid], ms);
  const float tot = blockReduceSum(sq, red, tid);
  if (tid == 0) atomicAdd(&scal[0], tot);
}

// ============================================================
// K7: aux decode + l2a / denom partial sums
// ============================================================
__global__ __launch_bounds__(256) void k_aux_decode(const float* __restrict__ resid,
                                                    const float* __restrict__ Wdec,
                                                    const float* __restrict__ musum,
                                                    const int* __restrict__ idxA, const float* __restrict__ valA,
                                                    const int* __restrict__ cntA,
                                                    float* __restrict__ scal) {
  __shared__ float red[256];
  const int b = blockIdx.x;
  const int tid = threadIdx.x;
  const int d0 = tid * 4;
  float a0 = 0.f, a1 = 0.f, a2 = 0.f, a3 = 0.f;
  const int c = cntA[b];
  const int* ip = idxA + (size_t)b * AUXK;
  const float* vp = valA + (size_t)b * AUXK;
  for (int e = 0; e < c; ++e) {
    const int id = ip[e];
    const float vv = vp[e];
    const float4 w = *(const float4*)(Wdec + (size_t)id * KD + d0);
    a0 += vv * w.x; a1 += vv * w.y; a2 += vv * w.z; a3 += vv * w.w;
  }
  const float acc[4] = {a0, a1, a2, a3};
  float l2 = 0.f, dn = 0.f;
#pragma unroll
  for (int j = 0; j < 4; ++j) {
    const int dim = d0 + j;
    const float r = resid[(size_t)b * KD + dim];
    const float t1 = r - acc[j];
    l2 += t1 * t1;
    const float mu = musum[dim & 255] * (1.0f / 16384.0f);
    const float t2 = r - mu;
    dn += t2 * t2;
  }
  const float l2t = blockReduceSum(l2, red, tid);
  if (tid == 0) atomicAdd(&scal[1], l2t);
  const float dnt = blockReduceSum(dn, red, tid);
  if (tid == 0) atomicAdd(&scal[2], dnt);
}

// ============================================================
// K8: scalar finalize
// ============================================================
__global__ void k_final(const float* __restrict__ scal, const int* __restrict__ scal_i,
                        float* __restrict__ out) {
  if (threadIdx.x == 0 && blockIdx.x == 0) {
    const float recon = scal[0] * (1.0f / 16384.0f);
    const float l2a   = scal[1] * (1.0f / 16384.0f);
    float den = scal[2] * (1.0f / 16384.0f);
    den = fmaxf(den, 1e-8f);
    float laux = l2a / den;
    if (laux != laux) laux = 0.0f;                                  // nan_to_num
    else if (laux > 3.4028235e38f) laux = 3.4028235e38f;
    else if (laux < -3.4028235e38f) laux = -3.4028235e38f;
    if (scal_i[3] <= 0) laux = 0.0f;                                // n_dead gate
    out[0] = recon + (1.0f / 32.0f) * laux;
  }
}

// ============================================================
extern "C" void kernel_launch(void* const* d_in, const int* in_sizes, int n_in,
                              void* d_out, int out_size, void* d_ws, size_t ws_size,
                              hipStream_t stream) {
  (void)in_sizes; (void)n_in; (void)out_size; (void)ws_size;
  const float* x    = (const float*)d_in[0];
  const float* Wenc = (const float*)d_in[1];
  const float* benc = (const float*)d_in[2];
  const float* Wdec = (const float*)d_in[3];
  const float* bdec = (const float*)d_in[4];
  const int*   ntsf = (const int*)d_in[5];

  float* out  = (float*)d_out;
  float* xhat = out + 1;
  float* z    = out + 1 + (size_t)BN * KD;  // z region doubles as `pre` scratch

  float* wsf = (float*)d_ws;
  int*   wsi = (int*)d_ws;
  float* scal   = wsf + OFF_SCAL;
  float* musum  = wsf + OFF_MUSUM;
  int*   active = wsi + OFF_ACTIVE;
  int*   dead   = wsi + OFF_DEAD;
  int*   cntM   = wsi + OFF_CNTM;
  int*   cntA   = wsi + OFF_CNTA;
  int*   idxM   = wsi + OFF_IDXM;
  float* valM   = wsf + OFF_VALM;
  int*   idxA   = wsi + OFF_IDXA;
  float* valA   = wsf + OFF_VALA;
  float* resid  = wsf + OFF_RES;

  k_zero<<<((int)ZERO_WORDS + 255) / 256, 256, 0, stream>>>(wsf, (int)ZERO_WORDS);
  k_encode<<<dim3(SN / 64, BN / 64), 256, 0, stream>>>(x, Wenc, benc, z);
  k_select_main<<<BN, 256, 0, stream>>>(z, idxM, valM, cntM, active);
  k_dead<<<(SN + 255) / 256, 256, 0, stream>>>(ntsf, active, dead, &wsi[3]);
  k_select_aux<<<BN, 256, 0, stream>>>(z, dead, idxM, valM, cntM, idxA, valA, cntA);
  k_decode<<<BN, 256, 0, stream>>>(x, Wdec, bdec, idxM, valM, cntM, xhat, resid);
  k_mu<<<64, 256, 0, stream>>>(resid, musum, scal);
  k_aux_decode<<<BN, 256, 0, stream>>>(resid, Wdec, musum, idxA, valA, cntA, scal);
  k_final<<<1, 1, 0, stream>>>(scal, wsi, out);
}